// DynamicRoutingCapsule_18657337934721
// MI455X (gfx1250) — compile-verified
//
#include <hip/hip_runtime.h>
#include <math.h>

typedef __attribute__((ext_vector_type(2))) float v2f;
typedef __attribute__((ext_vector_type(8))) float v8f;

// Problem sizes (compile-time)
constexpr int Bb = 32;    // batch
constexpr int Ss = 1024;  // sequence
constexpr int Dd = 256;   // input dim
constexpr int Nn = 32;    // capsules
constexpr int Cc = 64;    // capsule dim

// Workspace layout (float offsets)
constexpr size_t OFF_LOGITS = 0;                          // [B][S][N] 1,048,576
constexpr size_t OFF_CBUF   = OFF_LOGITS + (size_t)Bb*Ss*Nn;  // [B][S][N]
constexpr size_t OFF_XC     = OFF_CBUF   + (size_t)Bb*Ss*Nn;  // [B][N][D] 262,144
constexpr size_t OFF_WV     = OFF_XC     + (size_t)Bb*Nn*Dd;  // [B][N][D]
constexpr size_t OFF_S      = OFF_WV     + (size_t)Bb*Nn*Dd;  // [B][N][C] 65,536
constexpr size_t OFF_V      = OFF_S      + (size_t)Bb*Nn*Cc;  // [B][N][C]

__device__ __forceinline__ v8f wmma_f32_k4(v2f a, v2f b, v8f c) {
  // V_WMMA_F32_16X16X4_F32 : D = A(16x4) * B(4x16) + C(16x16), all f32
  return __builtin_amdgcn_wmma_f32_16x16x4_f32(false, a, false, b, (short)0, c,
                                               false, false);
}

// ---------------------------------------------------------------- zero logits
__global__ void caps_zero(float* __restrict__ p, int n) {
  int i = blockIdx.x * blockDim.x + threadIdx.x;
  if (i < n) p[i] = 0.0f;
}

// ----------------------------------------------------- softmax over capsules
// c[b,s,:] = softmax(logits[b,s,:]) along N=32
__global__ void caps_softmax(const float* __restrict__ logits,
                             float* __restrict__ cbuf) {
  int t = blockIdx.x * blockDim.x + threadIdx.x;  // 0 .. B*S-1
  const float* lg = logits + (size_t)t * Nn;
  float* co = cbuf + (size_t)t * Nn;
  float e[Nn];
  float m = lg[0];
#pragma unroll
  for (int n = 1; n < Nn; ++n) m = fmaxf(m, lg[n]);
  float sum = 0.0f;
#pragma unroll
  for (int n = 0; n < Nn; ++n) {
    e[n] = __expf(lg[n] - m);
    sum += e[n];
  }
  float inv = 1.0f / sum;
#pragma unroll
  for (int n = 0; n < Nn; ++n) co[n] = e[n] * inv;
}

// ------------------------------------------------- xc[b,n,d] = sum_s c * x
// Per b: (c_b)^T [N x S] * x_b [S x D].  1024 wave-tiles, K-loop = S/4.
__global__ void caps_xc_gemm(const float* __restrict__ x,
                             const float* __restrict__ cbuf,
                             float* __restrict__ xc) {
  int gw = blockIdx.x * (blockDim.x >> 5) + (threadIdx.x >> 5);
  int lane = threadIdx.x & 31;
  int b = gw >> 5;          // 32 tiles per batch
  int rem = gw & 31;
  int mtile = rem >> 4;     // capsule-row tile (0..1)
  int ntile = rem & 15;     // d-col tile (0..15)
  int half = lane >> 4;
  int l16 = lane & 15;
  int m = mtile * 16 + l16;      // capsule index (A row)
  int col = ntile * 16 + l16;    // d index (B/D col)
  const float* cb = cbuf + (size_t)b * Ss * Nn;  // [S][N]
  const float* xb = x + (size_t)b * Ss * Dd;     // [S][D]
  __builtin_prefetch(&xb[(size_t)(2 * half) * Dd + col], 0, 1);
  v8f acc = {};
  for (int k0 = 0; k0 < Ss; k0 += 4) {
    int k = k0 + 2 * half;
    v2f a, bv;
    a.x = cb[(size_t)k * Nn + m];
    a.y = cb[(size_t)(k + 1) * Nn + m];
    bv.x = xb[(size_t)k * Dd + col];
    bv.y = xb[(size_t)(k + 1) * Dd + col];
    acc = wmma_f32_k4(a, bv, acc);
  }
  float* out = xc + (size_t)b * Nn * Dd;
#pragma unroll
  for (int r = 0; r < 8; ++r) {
    int row = r + 8 * half;  // capsule index within tile
    out[(size_t)(mtile * 16 + row) * Dd + col] = acc[r];
  }
}

// --------------------------------------- s[b,n,c] = sum_d xc[b,n,d]*W[n,d,c]
// Per n: xc[:,n,:] [B x D] * W[n] [D x C].  256 wave-tiles, K-loop = D/4.
__global__ void caps_s_gemm(const float* __restrict__ xc,
                            const float* __restrict__ W,
                            float* __restrict__ sbuf) {
  int gw = blockIdx.x * (blockDim.x >> 5) + (threadIdx.x >> 5);
  int lane = threadIdx.x & 31;
  int n = gw >> 3;          // 8 tiles per capsule
  int rem = gw & 7;
  int mtile = rem >> 2;     // batch-row tile (0..1)
  int ntile = rem & 3;      // c-col tile (0..3)
  int half = lane >> 4;
  int l16 = lane & 15;
  int m = mtile * 16 + l16;    // batch index (A row)
  int col = ntile * 16 + l16;  // c index
  const float* Wn = W + (size_t)n * Dd * Cc;  // [D][C]
  v8f acc = {};
  for (int k0 = 0; k0 < Dd; k0 += 4) {
    int k = k0 + 2 * half;
    v2f a, bv;
    a.x = xc[(size_t)m * Nn * Dd + (size_t)n * Dd + k];
    a.y = xc[(size_t)m * Nn * Dd + (size_t)n * Dd + k + 1];
    bv.x = Wn[(size_t)k * Cc + col];
    bv.y = Wn[(size_t)(k + 1) * Cc + col];
    acc = wmma_f32_k4(a, bv, acc);
  }
#pragma unroll
  for (int r = 0; r < 8; ++r) {
    int row = r + 8 * half;  // batch index within tile
    sbuf[(size_t)(mtile * 16 + row) * Nn * Cc + (size_t)n * Cc + col] = acc[r];
  }
}

// ------------------------------------------------------------------- squash
// v = (|s|^2/(1+|s|^2)) * s / sqrt(|s|^2 + eps); one block per (b,n)
__global__ void caps_squash(const float* __restrict__ sbuf,
                            float* __restrict__ vbuf,
                            float* __restrict__ out) {
  int bn = blockIdx.x;     // b*N + n
  int c = threadIdx.x;     // 0..63
  float sv = sbuf[(size_t)bn * Cc + c];
  __shared__ float red[Cc];
  red[c] = sv * sv;
  __syncthreads();
#pragma unroll
  for (int off = Cc / 2; off >= 1; off >>= 1) {
    if (c < off) red[c] += red[c + off];
    __syncthreads();
  }
  float sq = red[0];
  float scale = (sq / (1.0f + sq)) * rsqrtf(sq + 1e-8f);
  float vv = scale * sv;
  vbuf[(size_t)bn * Cc + c] = vv;
  out[(size_t)bn * Cc + c] = vv;
}

// ------------------------------------ Wv[b,n,d] = sum_c W[n,d,c]*v[b,n,c]
// Per n: v[:,n,:] [B x C] * W[n]^T [C x D].  1024 wave-tiles, K-loop = C/4.
__global__ void caps_wv_gemm(const float* __restrict__ W,
                             const float* __restrict__ vbuf,
                             float* __restrict__ wv) {
  int gw = blockIdx.x * (blockDim.x >> 5) + (threadIdx.x >> 5);
  int lane = threadIdx.x & 31;
  int n = gw >> 5;          // 32 tiles per capsule
  int rem = gw & 31;
  int mtile = rem >> 4;     // batch-row tile (0..1)
  int ntile = rem & 15;     // d-col tile (0..15)
  int half = lane >> 4;
  int l16 = lane & 15;
  int m = mtile * 16 + l16;    // batch index
  int col = ntile * 16 + l16;  // d index
  const float* Wn = W + (size_t)n * Dd * Cc;  // [D][C]
  v8f acc = {};
  for (int k0 = 0; k0 < Cc; k0 += 4) {
    int k = k0 + 2 * half;   // k runs over c
    v2f a, bv;
    a.x = vbuf[(size_t)m * Nn * Cc + (size_t)n * Cc + k];
    a.y = vbuf[(size_t)m * Nn * Cc + (size_t)n * Cc + k + 1];
    bv.x = Wn[(size_t)col * Cc + k];        // W[n, d=col, c=k]
    bv.y = Wn[(size_t)col * Cc + k + 1];
    acc = wmma_f32_k4(a, bv, acc);
  }
#pragma unroll
  for (int r = 0; r < 8; ++r) {
    int row = r + 8 * half;
    wv[(size_t)(mtile * 16 + row) * Nn * Dd + (size_t)n * Dd + col] = acc[r];
  }
}

// --------------- logits[b,s,n] += x[b,s,:] . Wv[b,n,:]   (agreement update)
// Per b: x_b [S x D] * Wv_b^T [D x N].  4096 wave-tiles, K-loop = D/4.
__global__ void caps_agree_gemm(const float* __restrict__ x,
                                const float* __restrict__ wv,
                                float* __restrict__ logits) {
  int gw = blockIdx.x * (blockDim.x >> 5) + (threadIdx.x >> 5);
  int lane = threadIdx.x & 31;
  int b = gw >> 7;          // 128 tiles per batch
  int rem = gw & 127;
  int mtile = rem >> 1;     // s-row tile (0..63)
  int ntile = rem & 1;      // capsule-col tile (0..1)
  int half = lane >> 4;
  int l16 = lane & 15;
  int m = mtile * 16 + l16;    // s index (A row)
  int col = ntile * 16 + l16;  // capsule index (B col)
  const float* xb = x + (size_t)b * Ss * Dd;        // [S][D]
  const float* wb = wv + (size_t)b * Nn * Dd;       // [N][D]
  __builtin_prefetch(&xb[(size_t)m * Dd], 0, 1);
  v8f acc = {};
  for (int k0 = 0; k0 < Dd; k0 += 4) {
    int k = k0 + 2 * half;
    v2f a, bv;
    a.x = xb[(size_t)m * Dd + k];
    a.y = xb[(size_t)m * Dd + k + 1];
    bv.x = wb[(size_t)col * Dd + k];      // Wv[b, n=col, d=k]
    bv.y = wb[(size_t)col * Dd + k + 1];
    acc = wmma_f32_k4(a, bv, acc);
  }
  float* lg = logits + (size_t)b * Ss * Nn;
#pragma unroll
  for (int r = 0; r < 8; ++r) {
    int row = r + 8 * half;  // s index within tile
    size_t idx = (size_t)(mtile * 16 + row) * Nn + col;
    lg[idx] += acc[r];
  }
}

extern "C" void kernel_launch(void* const* d_in, const int* in_sizes, int n_in,
                              void* d_out, int out_size, void* d_ws,
                              size_t ws_size, hipStream_t stream) {
  const float* x = (const float*)d_in[0];  // [B,S,D]
  const float* W = (const float*)d_in[1];  // [N,D,C]
  float* out = (float*)d_out;              // [B,N,C]
  float* ws = (float*)d_ws;

  float* logits = ws + OFF_LOGITS;
  float* cbuf   = ws + OFF_CBUF;
  float* xc     = ws + OFF_XC;
  float* wv     = ws + OFF_WV;
  float* sbuf   = ws + OFF_S;
  float* vbuf   = ws + OFF_V;

  // logits = 0
  caps_zero<<<(Bb * Ss * Nn + 511) / 512, 512, 0, stream>>>(logits,
                                                            Bb * Ss * Nn);

  for (int iter = 0; iter < 3; ++iter) {
    // c = softmax(logits) over capsules
    caps_softmax<<<(Bb * Ss) / 256, 256, 0, stream>>>(logits, cbuf);
    // xc = einsum('bsn,bsd->bnd', c, x)      [1024 waves]
    caps_xc_gemm<<<256, 128, 0, stream>>>(x, cbuf, xc);
    // s = einsum('bnd,ndc->bnc', xc, W)      [256 waves]
    caps_s_gemm<<<64, 128, 0, stream>>>(xc, W, sbuf);
    // v = squash(s)  (also writes d_out; final iteration's v persists)
    caps_squash<<<Bb * Nn, Cc, 0, stream>>>(sbuf, vbuf, out);
    if (iter < 2) {
      // Wv = einsum('ndc,bnc->bnd', W, v)    [1024 waves]
      caps_wv_gemm<<<256, 128, 0, stream>>>(W, vbuf, wv);
      // logits += einsum('bsd,bnd->bsn', x, Wv)   [4096 waves]
      caps_agree_gemm<<<1024, 128, 0, stream>>>(x, wv, logits);
    }
  }
}